// SC2UnguidedDense_25409026523333
// MI455X (gfx1250) — compile-verified
//
#include <hip/hip_runtime.h>
#include <hip/hip_bf16.h>

typedef __attribute__((ext_vector_type(2))) float v2f;
typedef __attribute__((ext_vector_type(8))) float v8f;
typedef __attribute__((ext_vector_type(4))) unsigned v4u;
typedef __attribute__((ext_vector_type(8))) int v8i;
typedef __attribute__((ext_vector_type(4))) int v4i;
typedef __attribute__((address_space(3))) float lds_float;

#define EPSN 1e-20f

__device__ __forceinline__ unsigned rfl(unsigned x) {
  return (unsigned)__builtin_amdgcn_readfirstlane((int)x);
}

// ---------------------------------------------------------------------------
// TDM: DMA a 3-D tile (width x nrows x nchan fp32) from global into LDS.
// D# built per CDNA5 ISA 8.3-8.6: group0 = {count/lds_addr/global_addr/type},
// group1 = {data_size, tensor_dim0/1, tile_dim0/1/2, strides}, group2 =
// {tensor_dim2}. Tile is pre-clipped to the image so no OOB handling needed.
// Toolchain uses the 6-arg builtin: (v4u, v8i, v4i, v4i, v8i, i32 cpol).
// ---------------------------------------------------------------------------
__device__ __forceinline__ void tdm_load_tile(const float* gbase, unsigned lds_off,
                                              int width, int nrows, int nchan,
                                              int stride0, int stride1) {
  unsigned long long ga = (unsigned long long)gbase;
  unsigned w = (unsigned)width, h = (unsigned)nrows, cdim = (unsigned)nchan;
  unsigned s0 = (unsigned)stride0, s1 = (unsigned)stride1;
  v4u g0;
  g0.x = 1u;                                                    // count=1 valid user D#
  g0.y = rfl(lds_off);                                          // lds_addr (bytes)
  g0.z = rfl((unsigned)(ga & 0xffffffffull));                   // global_addr[31:0]
  g0.w = rfl((unsigned)((ga >> 32) & 0x1ffffffull)) | (2u << 30); // addr[56:32] | type=2
  v8i g1;
  g1[0] = (int)(2u << 16);                                      // data_size=4B, mask=0
  g1[1] = (int)rfl((w & 0xffffu) << 16);                        // tensor_dim0[15:0]
  g1[2] = (int)rfl((w >> 16) | ((h & 0xffffu) << 16));          // td0[31:16] | td1[15:0]
  g1[3] = (int)rfl((h >> 16) | (w << 16));                      // td1[31:16] | tile_dim0
  g1[4] = (int)rfl((h & 0xffffu) | (cdim << 16));               // tile_dim1 | tile_dim2
  g1[5] = (int)rfl(s0);                                         // dim0 stride [31:0]
  g1[6] = (int)rfl((s1 & 0xffffu) << 16);                       // s0[47:32]=0 | s1[15:0]
  g1[7] = (int)rfl(s1 >> 16);                                   // s1[47:16]
  v4i g2;
  g2[0] = (int)rfl(cdim);                                       // tensor_dim2
  g2[1] = 0; g2[2] = 0; g2[3] = 0;
  v4i g3 = {0, 0, 0, 0};
  v8i g4 = {0, 0, 0, 0, 0, 0, 0, 0};                            // unused extension
  __builtin_amdgcn_tensor_load_to_lds(g0, g1, g2, g3, g4, 0);
}

// ---------------------------------------------------------------------------
// Normalized convolution via V_WMMA_F32_16X16X4_F32 (exact fp32 tensor op).
//   D[pix, cout] = sum_k A[pix, k] * B[k, cout]
// Each wave owns a 16-wide x 8-row output tile; the input halo (CIN0 channels
// of conf and data) is DMA'd into a private LDS slice by the TDM, then the
// per-chunk A operands are gathered from LDS. Two WMMA accumulator chains
// share the weight operand: accC = conv(conf), accX = conv(x*conf).
// Concat+upsample fusion: CIN1 channels come from a half-resolution source
// read directly from global (L2-resident). UPFIRST selects concat order.
// ---------------------------------------------------------------------------
template<int CIN0, int CIN1, int KSZ, int COUT, int UPFIRST>
__global__ __launch_bounds__(256) void nconv_wmma(
    const float* __restrict__ x0s, const float* __restrict__ c0s,
    const float* __restrict__ x1s, const float* __restrict__ c1s,
    const float* __restrict__ w,   const float* __restrict__ bias,
    float* __restrict__ xo, float* __restrict__ co,
    int B, int H, int W)
{
  constexpr int CIN  = CIN0 + CIN1;
  constexpr int PAD  = (KSZ - 1) / 2;
  constexpr int KTOT = CIN * KSZ * KSZ;
  constexpr int NCH  = (KTOT + 3) / 4;
  constexpr int ROWS = 8;                         // output rows per tile
  constexpr int HR   = ROWS + KSZ - 1;            // halo rows
  constexpr int MAXW = 16 + KSZ - 1;              // halo cols
  constexpr int LDSF = CIN0 * HR * MAXW;          // floats per plane per wave

  __shared__ float sC[8][LDSF];
  __shared__ float sX[8][LDSF];

  const int lane  = threadIdx.x & 31;
  const int ncol  = lane & 15;                    // N column == A-row pixel id
  const int khalf = (lane >> 4) * 2;              // lanes 16..31: k-slots {2,3}
  const int wslot = threadIdx.x >> 5;

  const unsigned ldsC_off = (unsigned)(size_t)(lds_float*)(&sC[wslot][0]);
  const unsigned ldsX_off = (unsigned)(size_t)(lds_float*)(&sX[wslot][0]);
  const float* sCw = &sC[wslot][0];
  const float* sXw = &sX[wslot][0];

  // --- Preload weight (B-matrix) chunks into registers, zero-padded ---
  v2f Bm[NCH];
#pragma unroll
  for (int cc = 0; cc < NCH; ++cc) {
#pragma unroll
    for (int j = 0; j < 2; ++j) {
      int k = cc * 4 + khalf + j;
      float wv = 0.0f;
      if (ncol < COUT && k < KTOT) wv = w[ncol * KTOT + k];
      Bm[cc][j] = wv;
    }
  }
  float inv_s = 1.0f, bv = 0.0f;
  if (ncol < COUT) {
    float s = 0.0f;
    for (int k = 0; k < KTOT; ++k) s += w[ncol * KTOT + k];
    inv_s = 1.0f / s;
    bv = bias[ncol];
  }

  const int tpr    = W >> 4;
  const int trows  = H / ROWS;
  const int ntiles = B * trows * tpr;
  const int wpb    = blockDim.x >> 5;
  const int wave   = blockIdx.x * wpb + wslot;
  const int nwaves = gridDim.x * wpb;
  const int Hh = H >> 1, Wh = W >> 1;

  for (int t = wave; t < ntiles; t += nwaves) {   // wave-uniform: EXEC stays full
    int b  = t / (trows * tpr);
    int r  = t - b * trows * tpr;
    int y0 = (r / tpr) * ROWS;
    int xb = (r - (r / tpr) * tpr) << 4;

    // image-clipped halo rectangle (conv zero-pad handled by tap guard)
    int y_lo = y0 - PAD;        if (y_lo < 0) y_lo = 0;
    int y_hi = y0 + ROWS - 1 + PAD; if (y_hi > H - 1) y_hi = H - 1;
    int x_lo = xb - PAD;        if (x_lo < 0) x_lo = 0;
    int x_hi = xb + 15 + PAD;   if (x_hi > W - 1) x_hi = W - 1;
    int nrows = y_hi - y_lo + 1;
    int width = x_hi - x_lo + 1;

    // WAR: previous iteration's LDS reads must drain before TDM overwrites
    asm volatile("s_wait_dscnt 0x0" ::: "memory");
    size_t goff = ((size_t)(b * CIN0) * H + y_lo) * W + x_lo;
    tdm_load_tile(c0s + goff, ldsC_off, width, nrows, CIN0, W, H * W);
    tdm_load_tile(x0s + goff, ldsX_off, width, nrows, CIN0, W, H * W);
    __builtin_amdgcn_s_wait_tensorcnt(0);
    asm volatile("" ::: "memory");                // fence LDS reads below the wait

#pragma unroll
    for (int ry = 0; ry < ROWS; ++ry) {
      int y = y0 + ry;
      v8f accC = {0.f,0.f,0.f,0.f,0.f,0.f,0.f,0.f};
      v8f accX = {0.f,0.f,0.f,0.f,0.f,0.f,0.f,0.f};

#pragma unroll
      for (int cc = 0; cc < NCH; ++cc) {
        v2f Ac = {0.f, 0.f};
        v2f Ax = {0.f, 0.f};
#pragma unroll
        for (int j = 0; j < 2; ++j) {
          int k = cc * 4 + khalf + j;
          if (k < KTOT) {
            int ci = k / (KSZ * KSZ);
            int rr = k - ci * (KSZ * KSZ);
            int dy = rr / KSZ;
            int dx = rr - dy * KSZ;
            int iy = y + dy - PAD;
            int ix = xb + ncol + dx - PAD;
            if (iy >= y_lo && iy <= y_hi && ix >= x_lo && ix <= x_hi) {
              bool fromUp;
              int cloc;
              if (UPFIRST) { fromUp = (ci < CIN1); cloc = fromUp ? ci : (ci - CIN1); }
              else         { fromUp = (ci >= CIN0); cloc = fromUp ? (ci - CIN0) : ci; }
              float cv, xv;
              if (CIN1 > 0 && fromUp) {
                size_t idx = ((size_t)(b * CIN1 + cloc) * Hh + (iy >> 1)) * Wh + (ix >> 1);
                cv = c1s[idx]; xv = x1s[idx];
              } else {
                int li = (cloc * nrows + (iy - y_lo)) * width + (ix - x_lo);
                cv = sCw[li]; xv = sXw[li];
              }
              Ac[j] = cv;
              Ax[j] = xv * cv;
            }
          }
        }
        accC = __builtin_amdgcn_wmma_f32_16x16x4_f32(false, Ac, false, Bm[cc],
                                                     (short)0, accC, false, false);
        accX = __builtin_amdgcn_wmma_f32_16x16x4_f32(false, Ax, false, Bm[cc],
                                                     (short)0, accX, false, false);
      }

      if (ncol < COUT) {
        int px = xb + ((lane >> 4) << 3);
        size_t base = ((size_t)(b * COUT + ncol) * H + y) * W + px;
#pragma unroll
        for (int e = 0; e < 8; ++e) {
          float dC = accC[e];
          float dX = accX[e];
          xo[base + e] = dX / (dC + EPSN) + bv;
          co[base + e] = dC * inv_s;
        }
      }
    }
  }
}

// ---------------------------------------------------------------------------
// 2x2/stride-2 max-pool on conf, gathering data at argmax (first-max ties),
// c_ds = cmax/4.
// ---------------------------------------------------------------------------
__global__ __launch_bounds__(256) void maxpool_ds(
    const float* __restrict__ x, const float* __restrict__ c,
    float* __restrict__ xd, float* __restrict__ cd,
    int BC, int H, int W)
{
  int Ho = H >> 1, Wo = W >> 1;
  int N = BC * Ho * Wo;
  int i = blockIdx.x * blockDim.x + threadIdx.x;
  if (i >= N) return;
  int wo = i % Wo;
  int t  = i / Wo;
  int ho = t % Ho;
  int bc = t / Ho;
  const float* cb = c + (size_t)bc * H * W;
  const float* xb = x + (size_t)bc * H * W;
  int y = ho << 1, x0 = wo << 1;
  float c00 = cb[(size_t)y * W + x0];
  float c01 = cb[(size_t)y * W + x0 + 1];
  float c10 = cb[(size_t)(y + 1) * W + x0];
  float c11 = cb[(size_t)(y + 1) * W + x0 + 1];
  float cm = c00; int am = 0;
  if (c01 > cm) { cm = c01; am = 1; }
  if (c10 > cm) { cm = c10; am = 2; }
  if (c11 > cm) { cm = c11; am = 3; }
  float xv;
  if      (am == 0) xv = xb[(size_t)y * W + x0];
  else if (am == 1) xv = xb[(size_t)y * W + x0 + 1];
  else if (am == 2) xv = xb[(size_t)(y + 1) * W + x0];
  else              xv = xb[(size_t)(y + 1) * W + x0 + 1];
  xd[i] = xv;
  cd[i] = cm * 0.25f;
}

// ---------------------------------------------------------------------------
// Host-side orchestration
// ---------------------------------------------------------------------------
static inline int nconv_blocks(int Bn, int H, int W) {
  int ntiles = Bn * (H / 8) * (W >> 4);
  return (ntiles + 7) / 8;   // 8 waves / 256-thread block
}

extern "C" void kernel_launch(void* const* d_in, const int* in_sizes, int n_in,
                              void* d_out, int out_size, void* d_ws, size_t ws_size,
                              hipStream_t stream) {
  (void)in_sizes; (void)n_in; (void)out_size; (void)ws_size;

  const float* x0 = (const float*)d_in[0];
  const float* c0 = (const float*)d_in[1];
  const float* w1 = (const float*)d_in[2];  const float* b1 = (const float*)d_in[3];
  const float* w2 = (const float*)d_in[4];  const float* b2 = (const float*)d_in[5];
  const float* w3 = (const float*)d_in[6];  const float* b3 = (const float*)d_in[7];
  const float* w4 = (const float*)d_in[8];  const float* b4 = (const float*)d_in[9];
  const float* w5 = (const float*)d_in[10]; const float* b5 = (const float*)d_in[11];
  const float* w6 = (const float*)d_in[12]; const float* b6 = (const float*)d_in[13];
  const float* w7 = (const float*)d_in[14]; const float* b7 = (const float*)d_in[15];

  const int B = 8;
  const size_t S512 = (size_t)B * 2 * 512 * 512;
  const size_t S256 = (size_t)B * 2 * 256 * 256;
  const size_t S128 = (size_t)B * 2 * 128 * 128;
  const size_t S64  = (size_t)B * 2 * 64 * 64;

  float* ws = (float*)d_ws;
  size_t off = 0;
  float* P512x = ws + off; off += S512;  float* P512c = ws + off; off += S512;
  float* Q512x = ws + off; off += S512;  float* Q512c = ws + off; off += S512;
  float* X1x   = ws + off; off += S512;  float* X1c   = ws + off; off += S512;
  float* D256x = ws + off; off += S256;  float* D256c = ws + off; off += S256;
  float* P256x = ws + off; off += S256;  float* P256c = ws + off; off += S256;
  float* X2x   = ws + off; off += S256;  float* X2c   = ws + off; off += S256;
  float* D128x = ws + off; off += S128;  float* D128c = ws + off; off += S128;
  float* X3x   = ws + off; off += S128;  float* X3c   = ws + off; off += S128;
  float* D64x  = ws + off; off += S64;   float* D64c  = ws + off; off += S64;
  float* X4x   = ws + off; off += S64;   float* X4c   = ws + off; off += S64;

  dim3 blk(256);

  // encoder @512: L1(w1,1->2,5x5), L2(w2), L3(w3)
  nconv_wmma<1,0,5,2,0><<<nconv_blocks(B,512,512), blk, 0, stream>>>(
      x0, c0, x0, c0, w1, b1, P512x, P512c, B, 512, 512);
  nconv_wmma<2,0,5,2,0><<<nconv_blocks(B,512,512), blk, 0, stream>>>(
      P512x, P512c, P512x, P512c, w2, b2, Q512x, Q512c, B, 512, 512);
  nconv_wmma<2,0,5,2,0><<<nconv_blocks(B,512,512), blk, 0, stream>>>(
      Q512x, Q512c, Q512x, Q512c, w3, b3, X1x, X1c, B, 512, 512);

  // pool -> 256, L4(w2), L5(w3)
  {
    int N = B * 2 * 256 * 256;
    maxpool_ds<<<(N + 255) / 256, blk, 0, stream>>>(X1x, X1c, D256x, D256c, B * 2, 512, 512);
  }
  nconv_wmma<2,0,5,2,0><<<nconv_blocks(B,256,256), blk, 0, stream>>>(
      D256x, D256c, D256x, D256c, w2, b2, P256x, P256c, B, 256, 256);
  nconv_wmma<2,0,5,2,0><<<nconv_blocks(B,256,256), blk, 0, stream>>>(
      P256x, P256c, P256x, P256c, w3, b3, X2x, X2c, B, 256, 256);

  // pool -> 128, L6(w2)
  {
    int N = B * 2 * 128 * 128;
    maxpool_ds<<<(N + 255) / 256, blk, 0, stream>>>(X2x, X2c, D128x, D128c, B * 2, 256, 256);
  }
  nconv_wmma<2,0,5,2,0><<<nconv_blocks(B,128,128), blk, 0, stream>>>(
      D128x, D128c, D128x, D128c, w2, b2, X3x, X3c, B, 128, 128);

  // pool -> 64, L7(w2)
  {
    int N = B * 2 * 64 * 64;
    maxpool_ds<<<(N + 255) / 256, blk, 0, stream>>>(X3x, X3c, D64x, D64c, B * 2, 128, 128);
  }
  nconv_wmma<2,0,5,2,0><<<nconv_blocks(B,64,64), blk, 0, stream>>>(
      D64x, D64c, D64x, D64c, w2, b2, X4x, X4c, B, 64, 64);

  // decoder: L8 @128 concat(x3_ds, up(x4_ds)) w4 -> X34 (reuse D128)
  nconv_wmma<2,2,3,2,0><<<nconv_blocks(B,128,128), blk, 0, stream>>>(
      X3x, X3c, X4x, X4c, w4, b4, D128x, D128c, B, 128, 128);

  // L9 @256 concat(x2_ds, up(x34)) w5 -> X23 (reuse D256)
  nconv_wmma<2,2,3,2,0><<<nconv_blocks(B,256,256), blk, 0, stream>>>(
      X2x, X2c, D128x, D128c, w5, b5, D256x, D256c, B, 256, 256);

  // L10 @512 concat(up(x23), x1) w6 -> P512   (upsampled channels FIRST)
  nconv_wmma<2,2,3,2,1><<<nconv_blocks(B,512,512), blk, 0, stream>>>(
      X1x, X1c, D256x, D256c, w6, b6, P512x, P512c, B, 512, 512);

  // L11: 1x1 nconv (w7, 2->1) -> final xout / cout into d_out
  float* xout = (float*)d_out;
  float* cout = xout + (size_t)B * 1 * 512 * 512;
  nconv_wmma<2,0,1,1,0><<<nconv_blocks(B,512,512), blk, 0, stream>>>(
      P512x, P512c, P512x, P512c, w7, b7, xout, cout, B, 512, 512);
}